// SUSAGE_13056700580225
// MI455X (gfx1250) — compile-verified
//
#include <hip/hip_runtime.h>
#include <hip/hip_bf16.h>

typedef __attribute__((ext_vector_type(2))) float v2f;
typedef __attribute__((ext_vector_type(8))) float v8f;

// ---------------------------------------------------------------- utilities
__global__ void fill_zero_kernel(float* __restrict__ p, long long n) {
  long long i = (long long)blockIdx.x * blockDim.x + threadIdx.x;
  if (i < n) p[i] = 0.0f;
}

// cnt[dst] += 1 per edge
__global__ void count_kernel(const int* __restrict__ dst, float* __restrict__ cnt, int E) {
  int e = blockIdx.x * blockDim.x + threadIdx.x;
  if (e < E) atomicAdd(&cnt[dst[e]], 1.0f);
}

// agg[dst] += h[src]; one wave per edge, float4 per lane (128 floats / edge)
__global__ void scatter_kernel(const float* __restrict__ h, const int* __restrict__ src,
                               const int* __restrict__ dst, float* __restrict__ agg, int E) {
  long long t = (long long)blockIdx.x * blockDim.x + threadIdx.x;
  int e = (int)(t >> 5);
  int lane = (int)(t & 31);
  if (e >= E) return;
  int s = src[e];
  int d0 = dst[e];
  const float4 v = *((const float4*)(h + (size_t)s * 128) + lane);
  float* b = agg + (size_t)d0 * 128 + (size_t)lane * 4;
  atomicAdd(b + 0, v.x);
  atomicAdd(b + 1, v.y);
  atomicAdd(b + 2, v.z);
  atomicAdd(b + 3, v.w);
}

// ------------------------------------------------------- fused SAGE GEMM (WMMA)
// out[n][o] = maybe_relu( sum_k (A[n][k]/max(cnt[n],1))*Wl[o][k] + bl[o]
//                       + sum_k H[n][k]*Wr[o][k] )
// A (raw neighbor sums), H: [nrows,K] row-major; Wl,Wr: [O,K] row-major.
// One wave = one 16x16 output tile via V_WMMA_F32_16X16X4_F32 (full f32 precision).
// Mean-normalization is fused: each lane owns exactly one A row, so the divide
// collapses to one scalar reciprocal applied to the A fragment in registers.
// PARTIAL=1 handles O<16 (layer 2, O=8) with a branchless B-fragment mask;
// PARTIAL=0 (layers 0/1) feeds loads straight into v_wmma.
template <int RELU, int PARTIAL>
__global__ void sage_gemm_wmma_kernel(const float* __restrict__ A,
                                      const float* __restrict__ H,
                                      const float* __restrict__ cnt,
                                      const float* __restrict__ Wl,
                                      const float* __restrict__ bl,
                                      const float* __restrict__ Wr,
                                      float* __restrict__ out,
                                      int nrows, int K, int O, int ldo,
                                      int col_tiles, int total_tiles) {
  int wave = (int)(((long long)blockIdx.x * blockDim.x + threadIdx.x) >> 5);
  int lane = threadIdx.x & 31;
  if (wave >= total_tiles) return;  // wave-uniform: EXEC all-ones at every WMMA

  int row_tile = wave / col_tiles;
  int col_tile = wave - row_tile * col_tiles;
  int row0 = row_tile * 16;
  int o0   = col_tile * 16;
  int hi = lane >> 4;   // selects K pair (0: k,k+1 | 1: k+2,k+3)
  int lo = lane & 15;

  int arow = row0 + lo;
  if (arow >= nrows) arow = nrows - 1;      // clamp loads; stores are guarded
  int   ocol   = o0 + lo;
  bool  ovalid = PARTIAL ? (ocol < O) : true;
  int   ocolc  = ovalid ? ocol : 0;
  float omask  = ovalid ? 1.0f : 0.0f;

  float ainv = 1.0f / fmaxf(cnt[arow], 1.0f);   // fused segment-mean divide

  const float* Ap  = A  + (size_t)arow  * K + 2 * hi;
  const float* Hp  = H  + (size_t)arow  * K + 2 * hi;
  const float* Wlp = Wl + (size_t)ocolc * K + 2 * hi;
  const float* Wrp = Wr + (size_t)ocolc * K + 2 * hi;

  v8f acc0 = {};
  v8f acc1 = {};
#pragma unroll 2
  for (int k = 0; k < K; k += 8) {
    v2f a0 = *(const v2f*)(Ap + k) * ainv;       // A[row][k+2*hi..+1] / cnt
    v2f a1 = *(const v2f*)(Ap + k + 4) * ainv;
    v2f b0 = *(const v2f*)(Wlp + k);             // Wl[o][k+2*hi..+1] == B[k][o]
    v2f b1 = *(const v2f*)(Wlp + k + 4);
    if (PARTIAL) { b0 *= omask; b1 *= omask; }
    acc0 = __builtin_amdgcn_wmma_f32_16x16x4_f32(false, a0, false, b0,
                                                 (short)0, acc0, false, false);
    acc1 = __builtin_amdgcn_wmma_f32_16x16x4_f32(false, a1, false, b1,
                                                 (short)0, acc1, false, false);
  }
#pragma unroll 2
  for (int k = 0; k < K; k += 8) {
    v2f a0 = *(const v2f*)(Hp + k);
    v2f a1 = *(const v2f*)(Hp + k + 4);
    v2f b0 = *(const v2f*)(Wrp + k);
    v2f b1 = *(const v2f*)(Wrp + k + 4);
    if (PARTIAL) { b0 *= omask; b1 *= omask; }
    acc0 = __builtin_amdgcn_wmma_f32_16x16x4_f32(false, a0, false, b0,
                                                 (short)0, acc0, false, false);
    acc1 = __builtin_amdgcn_wmma_f32_16x16x4_f32(false, a1, false, b1,
                                                 (short)0, acc1, false, false);
  }
  v8f acc = acc0 + acc1;

  float bias = ovalid ? bl[ocolc] : 0.0f;
#pragma unroll
  for (int j = 0; j < 8; ++j) {        // D: VGPR j -> row row0 + j + 8*hi, col o0+lo
    int r = row0 + j + 8 * hi;
    float v = acc[j] + bias;
    if (RELU) v = fmaxf(v, 0.0f);
    if (ovalid && r < nrows) out[(size_t)r * ldo + ocol] = v;
  }
}

// ------------------------------------------------------------- log_softmax(8)
__global__ void logsoftmax8_kernel(const float* __restrict__ h, float* __restrict__ out, int N) {
  int n = blockIdx.x * blockDim.x + threadIdx.x;
  if (n >= N) return;
  float v[8];
#pragma unroll
  for (int j = 0; j < 8; ++j) v[j] = h[(size_t)n * 8 + j];
  float m = v[0];
#pragma unroll
  for (int j = 1; j < 8; ++j) m = fmaxf(m, v[j]);
  float s = 0.0f;
#pragma unroll
  for (int j = 0; j < 8; ++j) s += expf(v[j] - m);
  float lse = logf(s) + m;
#pragma unroll
  for (int j = 0; j < 8; ++j) out[(size_t)n * 8 + j] = v[j] - lse;
}

// ------------------------------------------------------------------ launcher
extern "C" void kernel_launch(void* const* d_in, const int* in_sizes, int n_in,
                              void* d_out, int out_size, void* d_ws, size_t ws_size,
                              hipStream_t stream) {
  const float* x   = (const float*)d_in[0];
  const int*   ei  = (const int*)d_in[1];   // JAX w/o x64 -> int32 [2,E]
  const float* Wl0 = (const float*)d_in[2];
  const float* bl0 = (const float*)d_in[3];
  const float* Wr0 = (const float*)d_in[4];
  const float* Wl1 = (const float*)d_in[5];
  const float* bl1 = (const float*)d_in[6];
  const float* Wr1 = (const float*)d_in[7];
  const float* Wl2 = (const float*)d_in[8];
  const float* bl2 = (const float*)d_in[9];
  const float* Wr2 = (const float*)d_in[10];

  const int d = 128, dout = 8;
  const int N = in_sizes[0] / d;
  const int E = in_sizes[1] / 2;
  const int* src = ei;
  const int* dst = ei + E;

  float* agg = (float*)d_ws;                 // [N,128]
  float* cnt = agg + (size_t)N * d;          // [N]   (contiguous after agg)
  float* h1  = cnt + N;                      // [N,128]
  float* h2  = h1 + (size_t)N * d;           // [N,128]
  float* hout = (float*)d_out;               // [N,8]
  float* lsm  = hout + (size_t)N * dout;     // [N,8]

  const long long aggN = (long long)N * d;
  dim3 blk(256);
  auto nblk = [](long long n) { return dim3((unsigned)((n + 255) / 256)); };

  const int row_tiles = (N + 15) / 16;  // 6250 exactly for N=100000
  const int ct = d / 16;                // 8 column tiles for O=128
  const int tt = row_tiles * ct;

  // ---- layer 0: x -> h1 (relu).  Zero agg+cnt in one pass (contiguous).
  fill_zero_kernel<<<nblk(aggN + N), blk, 0, stream>>>(agg, aggN + N);
  count_kernel<<<nblk(E), blk, 0, stream>>>(dst, cnt, E);
  scatter_kernel<<<nblk((long long)E * 32), blk, 0, stream>>>(x, src, dst, agg, E);
  sage_gemm_wmma_kernel<1, 0><<<nblk((long long)tt * 32), blk, 0, stream>>>(
      agg, x, cnt, Wl0, bl0, Wr0, h1, N, d, d, d, ct, tt);

  // ---- layer 1: h1 -> h2 (relu)
  fill_zero_kernel<<<nblk(aggN), blk, 0, stream>>>(agg, aggN);
  scatter_kernel<<<nblk((long long)E * 32), blk, 0, stream>>>(h1, src, dst, agg, E);
  sage_gemm_wmma_kernel<1, 0><<<nblk((long long)tt * 32), blk, 0, stream>>>(
      agg, h1, cnt, Wl1, bl1, Wr1, h2, N, d, d, d, ct, tt);

  // ---- layer 2: h2 -> hout (no relu, O=8 -> PARTIAL tile)
  fill_zero_kernel<<<nblk(aggN), blk, 0, stream>>>(agg, aggN);
  scatter_kernel<<<nblk((long long)E * 32), blk, 0, stream>>>(h2, src, dst, agg, E);
  sage_gemm_wmma_kernel<0, 1><<<nblk((long long)row_tiles * 32), blk, 0, stream>>>(
      agg, h2, cnt, Wl2, bl2, Wr2, hout, N, d, dout, dout, 1, row_tiles);

  // ---- log_softmax over 8 classes
  logsoftmax8_kernel<<<nblk(N), blk, 0, stream>>>(hout, lsm, N);
}